// MHESA_5368709120672
// MI455X (gfx1250) — compile-verified
//
#include <hip/hip_runtime.h>

// ---------------------------------------------------------------------------
// MHESA on MI455X (gfx1250): GEMM(WMMA f32) -> chunked EMA scan -> GEMM(WMMA)
//   B=8, N=4096, DIM=512, HEADS=8, DIM_HEAD=32, INNER=256
// Memory-bound (~190MB traffic, ~17 GFLOP) => full fp32 via
// V_WMMA_F32_16X16X4_F32. Weights pre-transposed once so both tiles are
// row-major copies; staged to LDS double-buffered via
// GLOBAL_LOAD_ASYNC_TO_LDS_B128 (ASYNCcnt) when available, overlapping the
// DMA of chunk i+1 with the 64 WMMAs of chunk i.
// ---------------------------------------------------------------------------

typedef float v2f __attribute__((ext_vector_type(2)));
typedef float v8f __attribute__((ext_vector_type(8)));
typedef int   v4i __attribute__((vector_size(16)));     // b128 payload type

typedef __attribute__((address_space(1))) v4i GV4i;     // global int4
typedef __attribute__((address_space(3))) v4i LV4i;     // LDS int4

#if __has_builtin(__builtin_amdgcn_global_load_async_to_lds_b128) && \
    __has_builtin(__builtin_amdgcn_s_wait_asynccnt)
#define MHESA_ASYNC 1
#else
#define MHESA_ASYNC 0
#endif

#define BATCH     8
#define NSEQ      4096
#define DIM_      512
#define INNER     256
#define NCHUNK    32
#define CHUNKLEN  128   // NCHUNK * CHUNKLEN == NSEQ

// ---------------------------------------------------------------------------
// One-time weight transpose: Wt[n][k] = W[k][n]  (W: K x Nout). 0.5 MB total,
// cost is noise; makes the GEMM's W-tile a contiguous row-major copy.
// ---------------------------------------------------------------------------
__global__ __launch_bounds__(256) void mhesa_transpose(
    const float* __restrict__ W, float* __restrict__ Wt, int K, int Nout)
{
    __shared__ float tile[16][17];
    const int kb = blockIdx.x * 16, nb = blockIdx.y * 16;
    const int tx = threadIdx.x & 15, ty = threadIdx.x >> 4;   // 16x16
    tile[ty][tx] = W[(size_t)(kb + ty) * Nout + nb + tx];
    __syncthreads();
    Wt[(size_t)(nb + ty) * K + kb + tx] = tile[tx][ty];
}

// ---------------------------------------------------------------------------
// fp32 WMMA GEMM:  C[M x Nout] = A[M x K] @ Wt[Nout x K]^T + bias[Nout]
// Block: 256 threads (8 waves), tile 64(M) x 256(N).
// Wave w: m-group = w%4 (16 rows), n-group = w/4 (128 cols = 8 WMMA tiles).
// KC=32 k-chunks, double-buffered LDS; A and Wt tiles both row-major in LDS
// (stride 36 floats = 144B: 16B-aligned for b128, bank-conflict-free reads);
// every fragment is one ds_load_b64 per lane.
// ---------------------------------------------------------------------------
template <int K>
__global__ __launch_bounds__(256) void mhesa_gemm_wmma(
    const float* __restrict__ A, const float* __restrict__ Wt,
    const float* __restrict__ bias, float* __restrict__ C, int Nout)
{
    constexpr int KC  = 32;
    constexpr int LDA = KC + 4;            // 36 floats = 144B rows
    constexpr int LDW = KC + 4;
    __shared__ float As[2][64 * LDA];      // 2 x  9.2 KB
    __shared__ float Ws[2][256 * LDW];     // 2 x 36.9 KB   (total ~92 KB)

    const int tid  = threadIdx.x;
    const int m0   = blockIdx.x * 64;
    const int n0   = blockIdx.y * 256;

    const int wave  = tid >> 5;            // wave32
    const int lane  = tid & 31;
    const int mg    = wave & 3;            // 4 m-groups of 16 rows
    const int ng    = wave >> 2;           // 2 n-groups of 128 cols
    const int lrow  = lane & 15;
    const int khalf = lane >> 4;           // frag K pair = (l/16)*2 + {0,1}
    const int wcol  = ng * 128 + lrow;     // + t*16 per tile

    v8f acc[8];
#pragma unroll
    for (int t = 0; t < 8; ++t)
        acc[t] = (v8f){0.f, 0.f, 0.f, 0.f, 0.f, 0.f, 0.f, 0.f};

    // Stage one KC-chunk (A: 64xKC, Wt: 256xKC) into LDS buffer `buf`.
    auto stage = [&](int kc, int buf) {
#pragma unroll
        for (int i = 0; i < 2; ++i) {                    // A tile: 512 float4
            const int idx = tid + i * 256;
            const int row = idx >> 3, c4 = idx & 7;
            const float* src = A + (size_t)(m0 + row) * K + kc + c4 * 4;
            float*       dst = &As[buf][row * LDA + c4 * 4];
#if MHESA_ASYNC
            __builtin_amdgcn_global_load_async_to_lds_b128(
                (GV4i*)src, (LV4i*)dst, 0, 0);
#else
            const float4 v = *(const float4*)src;
            dst[0] = v.x; dst[1] = v.y; dst[2] = v.z; dst[3] = v.w;
#endif
        }
#pragma unroll
        for (int i = 0; i < 8; ++i) {                    // Wt tile: 2048 float4
            const int idx = tid + i * 256;
            const int row = idx >> 3, c4 = idx & 7;
            const float* src = Wt + (size_t)(n0 + row) * K + kc + c4 * 4;
            float*       dst = &Ws[buf][row * LDW + c4 * 4];
#if MHESA_ASYNC
            __builtin_amdgcn_global_load_async_to_lds_b128(
                (GV4i*)src, (LV4i*)dst, 0, 0);
#else
            const float4 v = *(const float4*)src;
            dst[0] = v.x; dst[1] = v.y; dst[2] = v.z; dst[3] = v.w;
#endif
        }
    };

    stage(0, 0);
#if MHESA_ASYNC
    __builtin_amdgcn_s_wait_asynccnt(0);
#endif
    __syncthreads();

    int cur = 0;
    for (int kc = 0; kc < K; kc += KC) {
        // Kick off DMA for the next chunk into the other buffer (its previous
        // readers finished before the barrier that ended the last iteration).
        if (kc + KC < K) stage(kc + KC, cur ^ 1);

        // Compute on current buffer: 8 k-steps, 9 ds_load_b64 + 8 WMMA each.
        const float* a_base = &As[cur][(mg * 16 + lrow) * LDA + khalf * 2];
        const float* w_base = &Ws[cur][wcol * LDW + khalf * 2];
#pragma unroll
        for (int k0 = 0; k0 < KC; k0 += 4) {
            const v2f a = *(const v2f*)(a_base + k0);
            v2f bf[8];
#pragma unroll
            for (int t = 0; t < 8; ++t)
                bf[t] = *(const v2f*)(w_base + (t * 16) * LDW + k0);
#pragma unroll
            for (int t = 0; t < 8; ++t)
                acc[t] = __builtin_amdgcn_wmma_f32_16x16x4_f32(
                    false, a, false, bf[t], (short)0, acc[t], false, false);
        }

#if MHESA_ASYNC
        __builtin_amdgcn_s_wait_asynccnt(0);   // next buffer DMA complete
#endif
        __syncthreads();                        // all reads/writes settled
        cur ^= 1;
    }

    // D layout: VGPR i -> rows {i, i+8} (lane halves), col = lane%16.
    const int row0 = m0 + mg * 16 + khalf * 8;
    const int col0 = n0 + ng * 128 + lrow;
#pragma unroll
    for (int t = 0; t < 8; ++t) {
        const float bv = bias[col0 + t * 16];
        float* outp = C + (size_t)row0 * Nout + col0 + t * 16;
#pragma unroll
        for (int i = 0; i < 8; ++i)
            outp[(size_t)i * Nout] = acc[t][i] + bv;
    }
}

// ---------------------------------------------------------------------------
// EMA scan, chunk-parallel. Recurrence per channel c (h = c/32):
//   s_t = beta*s_{t-1} + alpha*(xp_t - xp_{t-1}),  s_{-1}=xp_{-1}=init[c]
// Pass1: per-chunk particular solution (s_in = 0) -> P[b][chunk][c]
// Pass2: carry scan over chunks with multiplier beta^CHUNKLEN -> Sin
// Pass3: replay chunk with correct incoming state, write y.
// ---------------------------------------------------------------------------
__device__ __forceinline__ float mhesa_alpha(const float* ap, int c) {
    return 1.0f / (1.0f + __expf(-ap[c >> 5]));
}

__global__ __launch_bounds__(256) void mhesa_scan_pass1(
    const float* __restrict__ xp, const float* __restrict__ alpha_param,
    const float* __restrict__ init_state, float* __restrict__ P)
{
    const int b  = blockIdx.x;
    const int ci = blockIdx.y;
    const int c  = threadIdx.x;
    const float alpha = mhesa_alpha(alpha_param, c);
    const float beta  = 1.0f - alpha;
    const int   t0    = ci * CHUNKLEN;
    const float* xrow = xp + ((size_t)b * NSEQ + t0) * INNER + c;
    float prev = (t0 == 0) ? init_state[c] : xrow[-INNER];
    float s = 0.0f;
    for (int t = 0; t < CHUNKLEN; ++t) {
        const float v = xrow[(size_t)t * INNER];
        s = beta * s + alpha * (v - prev);
        prev = v;
    }
    P[((size_t)b * NCHUNK + ci) * INNER + c] = s;
}

__global__ __launch_bounds__(256) void mhesa_scan_pass2(
    const float* __restrict__ P, const float* __restrict__ alpha_param,
    const float* __restrict__ init_state, float* __restrict__ Sin)
{
    const int b = blockIdx.x;
    const int c = threadIdx.x;
    const float beta = 1.0f - mhesa_alpha(alpha_param, c);
    float bCL = 1.0f;
    for (int i = 0; i < CHUNKLEN; ++i) bCL *= beta;   // beta^CHUNKLEN, exact
    float s = init_state[c];                           // incoming state chunk 0
    for (int ci = 0; ci < NCHUNK; ++ci) {
        const size_t idx = ((size_t)b * NCHUNK + ci) * INNER + c;
        Sin[idx] = s;
        s = P[idx] + bCL * s;
    }
}

__global__ __launch_bounds__(256) void mhesa_scan_pass3(
    const float* __restrict__ xp, const float* __restrict__ alpha_param,
    const float* __restrict__ init_state, const float* __restrict__ Sin,
    float* __restrict__ y)
{
    const int b  = blockIdx.x;
    const int ci = blockIdx.y;
    const int c  = threadIdx.x;
    const float alpha = mhesa_alpha(alpha_param, c);
    const float beta  = 1.0f - alpha;
    const int   t0    = ci * CHUNKLEN;
    const float* xrow = xp + ((size_t)b * NSEQ + t0) * INNER + c;
    float*       yrow = y  + ((size_t)b * NSEQ + t0) * INNER + c;
    float prev = (t0 == 0) ? init_state[c] : xrow[-INNER];
    float s = Sin[((size_t)b * NCHUNK + ci) * INNER + c];
    for (int t = 0; t < CHUNKLEN; ++t) {
        const float v = xrow[(size_t)t * INNER];
        s = beta * s + alpha * (v - prev);
        prev = v;
        yrow[(size_t)t * INNER] = s;
    }
}

// ---------------------------------------------------------------------------
extern "C" void kernel_launch(void* const* d_in, const int* in_sizes, int n_in,
                              void* d_out, int out_size, void* d_ws, size_t ws_size,
                              hipStream_t stream)
{
    const float* x           = (const float*)d_in[0];
    const float* W_in        = (const float*)d_in[1];
    const float* b_in        = (const float*)d_in[2];
    const float* W_out       = (const float*)d_in[3];
    const float* b_out       = (const float*)d_in[4];
    const float* init_state  = (const float*)d_in[5];
    const float* alpha_param = (const float*)d_in[6];
    float* out = (float*)d_out;

    const int M = BATCH * NSEQ;                       // 32768 rows

    // Workspace carve-up (all fully written before read):
    float* xp  = (float*)d_ws;                        // M*INNER
    float* y   = xp  + (size_t)M * INNER;             // M*INNER
    float* P   = y   + (size_t)M * INNER;             // BATCH*NCHUNK*INNER
    float* Sin = P   + (size_t)BATCH * NCHUNK * INNER;
    float* Wt1 = Sin + (size_t)BATCH * NCHUNK * INNER; // [INNER][DIM_]
    float* Wt2 = Wt1 + (size_t)INNER * DIM_;           // [DIM_][INNER]

    // 0) transpose weights once (tiny)
    mhesa_transpose<<<dim3(DIM_ / 16, INNER / 16), 256, 0, stream>>>(
        W_in, Wt1, DIM_, INNER);
    mhesa_transpose<<<dim3(INNER / 16, DIM_ / 16), 256, 0, stream>>>(
        W_out, Wt2, INNER, DIM_);

    // 1) xp = x @ W_in + b_in   (K=512, Nout=256)
    mhesa_gemm_wmma<DIM_><<<dim3(M / 64, 1), 256, 0, stream>>>(
        x, Wt1, b_in, xp, INNER);

    // 2) chunk-parallel EMA scan
    mhesa_scan_pass1<<<dim3(BATCH, NCHUNK), 256, 0, stream>>>(
        xp, alpha_param, init_state, P);
    mhesa_scan_pass2<<<dim3(BATCH), 256, 0, stream>>>(
        P, alpha_param, init_state, Sin);
    mhesa_scan_pass3<<<dim3(BATCH, NCHUNK), 256, 0, stream>>>(
        xp, alpha_param, init_state, Sin, y);

    // 3) out = y @ W_out + b_out   (K=256, Nout=512)
    mhesa_gemm_wmma<INNER><<<dim3(M / 64, 2), 256, 0, stream>>>(
        y, Wt2, b_out, out, DIM_);
}